// VideoThingGenerator_36928128811313
// MI455X (gfx1250) — compile-verified
//
#include <hip/hip_runtime.h>
#include <math.h>

typedef __attribute__((ext_vector_type(2))) float v2f;
typedef __attribute__((ext_vector_type(8))) float v8f;

#define B_N    2
#define F_NUM  4
#define NB     (B_N * F_NUM)      // 8
#define C_IN   256
#define C_OUT  64
#define N_PT   50
#define HW     67200              // 200*336
#define K_ROWS (N_PT * F_NUM)     // 200
#define PIX_PER_BLOCK 128         // 8 waves * 16 pixels; 67200/128 = 525 exactly
#define A_STRIDE 68               // 64 + 4 pad floats -> conflict-free ds_load_b64

// ---------------------------------------------------------------------------
// Kernel 1: mw / "kernels" = idx_feat^T @ weight^T + bias  -> ws[(n*50+nn)*64+d]
// Tiny (13 MFLOP) -- plain VALU, heavy cache reuse.
// ---------------------------------------------------------------------------
__launch_bounds__(256)
__global__ void mw_kernel(const float* __restrict__ idx_feat,
                          const float* __restrict__ weight,
                          const float* __restrict__ bias,
                          float* __restrict__ kw) {
  const int n = blockIdx.x;      // 0..7
  const int tid = threadIdx.x;
  for (int o = tid; o < N_PT * C_OUT; o += 256) {
    const int nn = o >> 6;       // 0..49
    const int d  = o & 63;       // 0..63  (consecutive tids -> consecutive d)
    const float* fp = idx_feat + (size_t)n * C_IN * N_PT + nn; // stride N_PT
    const float* wp = weight + (size_t)d * C_IN;
    float s = 0.0f;
    for (int c = 0; c < C_IN; ++c)
      s = fmaf(fp[(size_t)c * N_PT], wp[c], s);
    kw[((size_t)n * N_PT + nn) * C_OUT + d] = s + bias[d];
  }
}

// ---------------------------------------------------------------------------
// threefry2x32-20 (deterministic stand-in for jax.random.randint(key(42),...))
// ---------------------------------------------------------------------------
__device__ __forceinline__ unsigned rotl32(unsigned x, int r) {
  return (x << r) | (x >> (32 - r));
}
__device__ unsigned threefry2x32(unsigned k0, unsigned k1, unsigned c0, unsigned c1) {
  const unsigned ks2 = 0x1BD11BDAu ^ k0 ^ k1;
  unsigned x0 = c0 + k0, x1 = c1 + k1;
  const int rA[4] = {13, 15, 26, 6};
  const int rB[4] = {17, 29, 16, 24};
#pragma unroll
  for (int i = 0; i < 4; ++i) { x0 += x1; x1 = rotl32(x1, rA[i]); x1 ^= x0; }
  x0 += k1;  x1 += ks2 + 1u;
#pragma unroll
  for (int i = 0; i < 4; ++i) { x0 += x1; x1 = rotl32(x1, rB[i]); x1 ^= x0; }
  x0 += ks2; x1 += k0 + 2u;
#pragma unroll
  for (int i = 0; i < 4; ++i) { x0 += x1; x1 = rotl32(x1, rA[i]); x1 ^= x0; }
  x0 += k0;  x1 += k1 + 3u;
#pragma unroll
  for (int i = 0; i < 4; ++i) { x0 += x1; x1 = rotl32(x1, rB[i]); x1 ^= x0; }
  x0 += k1;  x1 += ks2 + 4u;
#pragma unroll
  for (int i = 0; i < 4; ++i) { x0 += x1; x1 = rotl32(x1, rA[i]); x1 ^= x0; }
  x0 += ks2; x1 += k0 + 5u;
  return x0 ^ x1;
}

// ---------------------------------------------------------------------------
// Kernel 2: contrastive loss. One workgroup; m(200,64) lives in LDS.
// loss = sum_b mean_k ( lse(logits_k) - pos_k/0.07 )
// negatives(k) = {k' : k'//4 != k//4} (196 cols, ascending) + 3 pads of last col
// pos_col from randint draw (threefry surrogate).
// ---------------------------------------------------------------------------
__launch_bounds__(256)
__global__ void con_loss_kernel(const float* __restrict__ kw,
                                float* __restrict__ loss_out) {
  __shared__ float m[K_ROWS * C_OUT];  // 51200 B
  __shared__ float red[256];
  const int tid = threadIdx.x;
  float total = 0.0f;

  for (int b = 0; b < B_N; ++b) {
    __syncthreads();
    // m[k = nn*4 + f][c] = kernels[b*4+f][nn][c]
    for (int idx = tid; idx < K_ROWS * C_OUT; idx += 256) {
      const int k = idx >> 6, c = idx & 63;
      const int f = k & 3, nn = k >> 2;
      m[idx] = kw[(((size_t)(b * F_NUM + f)) * N_PT + nn) * C_OUT + c];
    }
    __syncthreads();
    // row-wise L2 normalize
    if (tid < K_ROWS) {
      float ss = 0.0f;
      for (int c = 0; c < C_OUT; ++c) { float v = m[tid * C_OUT + c]; ss = fmaf(v, v, ss); }
      const float inv = 1.0f / fmaxf(sqrtf(ss), 1e-12f);
      for (int c = 0; c < C_OUT; ++c) m[tid * C_OUT + c] *= inv;
    }
    __syncthreads();

    float contrib = 0.0f;
    if (tid < K_ROWS) {
      const int k   = tid;
      const int lab = k >> 2;
      const int fo  = k & 3;
      const unsigned rb = threefry2x32(0u, 42u, (unsigned)b, (unsigned)k);
      const int r = (int)(rb % 3u);
      const int pos_off = (r >= fo) ? r + 1 : r;
      const int pos_col = lab * 4 + pos_off;
      // last negative column for this label (pad source): 199, or 195 if lab==49
      const int padcol = (lab == N_PT - 1) ? (K_ROWS - F_NUM - 1) : (K_ROWS - 1);

      float spos = 0.0f;
      for (int c = 0; c < C_OUT; ++c)
        spos = fmaf(m[k * C_OUT + c], m[pos_col * C_OUT + c], spos);
      spos *= (1.0f / 0.07f);

      // online logsumexp over {spos} U {negatives} then +3 pad duplicates
      float mrun = spos, srun = 1.0f, spad = 0.0f;
      for (int kp = 0; kp < K_ROWS; ++kp) {
        if ((kp >> 2) == lab) continue;
        float d = 0.0f;
        for (int c = 0; c < C_OUT; ++c)
          d = fmaf(m[k * C_OUT + c], m[kp * C_OUT + c], d);
        d *= (1.0f / 0.07f);
        if (kp == padcol) spad = d;
        if (d > mrun) { srun = srun * expf(mrun - d) + 1.0f; mrun = d; }
        else          { srun += expf(d - mrun); }
      }
      srun += 3.0f * expf(spad - mrun);   // pad appears 4x total (1 + 3)
      contrib = (mrun + logf(srun)) - spos;
    }
    red[tid] = contrib;
    __syncthreads();
    for (int s = 128; s > 0; s >>= 1) {
      if (tid < s) red[tid] += red[tid + s];
      __syncthreads();
    }
    if (tid == 0) total += red[0] * (1.0f / (float)K_ROWS);
  }
  if (tid == 0) loss_out[0] = total;
}

// ---------------------------------------------------------------------------
// Kernel 3: inst_pred[b][row][p] = sum_c kernels[b][row][c] * x[b][c][p]
// Per-batch GEMM M=50(pad 64), K=64, N=67200. HBM-bound: x read exactly once.
// WMMA: V_WMMA_F32_16X16X4_F32, 64 per wave (4 M-tiles x 16 K-steps).
// ---------------------------------------------------------------------------
__launch_bounds__(256)
__global__ void inst_pred_wmma(const float* __restrict__ x,
                               const float* __restrict__ kw,
                               float* __restrict__ out) {
  __shared__ float Al[64 * A_STRIDE];   // 17408 B, zero-padded 64x64 A tile
  const int b   = blockIdx.y;                 // 0..7
  const int p0  = blockIdx.x * PIX_PER_BLOCK; // pixel base
  const int tid = threadIdx.x;

  // Stage A = kernels[b] (50x64), rows 50..63 zeroed.
  for (int idx = tid; idx < 64 * 64; idx += 256) {
    const int r = idx >> 6, c = idx & 63;
    Al[r * A_STRIDE + c] = (r < N_PT)
        ? kw[((size_t)(b * N_PT + r)) * C_OUT + c] : 0.0f;
  }
  __syncthreads();

  const int wave  = tid >> 5;
  const int lane  = tid & 31;
  const int nloc  = lane & 15;   // N position within 16x16 tile
  const int khalf = lane >> 4;   // lane-half selects K pair / M+8 rows
  const int p     = p0 + wave * 16 + nloc;

  // B: 64(K) x 16(N) tile of x, ISA layout: lane holds K = kk*4 + 2*khalf (+1)
  const float* xb = x + (size_t)b * C_OUT * HW + p;
  v2f Bv[16];
#pragma unroll
  for (int kk = 0; kk < 16; ++kk) {
    const int c0 = kk * 4 + khalf * 2;
    v2f t;
    t.x = xb[(size_t)c0 * HW];
    t.y = xb[(size_t)(c0 + 1) * HW];
    Bv[kk] = t;
  }

  v8f acc0 = {}, acc1 = {}, acc2 = {}, acc3 = {};
  // A layout (32-bit 16x4): lane holds row M = nloc, K pair = kk*4 + 2*khalf
  const float* Ab = Al + nloc * A_STRIDE + khalf * 2;
#pragma unroll
  for (int kk = 0; kk < 16; ++kk) {
    const v2f a0 = *(const v2f*)(Ab + 0 * 16 * A_STRIDE + kk * 4);
    const v2f a1 = *(const v2f*)(Ab + 1 * 16 * A_STRIDE + kk * 4);
    const v2f a2 = *(const v2f*)(Ab + 2 * 16 * A_STRIDE + kk * 4);
    const v2f a3 = *(const v2f*)(Ab + 3 * 16 * A_STRIDE + kk * 4);
    // 4 independent accumulator chains interleaved to hide WMMA RAW latency
    acc0 = __builtin_amdgcn_wmma_f32_16x16x4_f32(false, a0, false, Bv[kk],
                                                 (short)0, acc0, false, false);
    acc1 = __builtin_amdgcn_wmma_f32_16x16x4_f32(false, a1, false, Bv[kk],
                                                 (short)0, acc1, false, false);
    acc2 = __builtin_amdgcn_wmma_f32_16x16x4_f32(false, a2, false, Bv[kk],
                                                 (short)0, acc2, false, false);
    acc3 = __builtin_amdgcn_wmma_f32_16x16x4_f32(false, a3, false, Bv[kk],
                                                 (short)0, acc3, false, false);
  }

  // D layout: VGPR j, lane-half h -> row = 16*mt + 8*h + j, col = nloc
  float* ob = out + (size_t)b * N_PT * HW + p;
#define STORE_TILE(ACC, MT)                                            \
  _Pragma("unroll")                                                    \
  for (int j = 0; j < 8; ++j) {                                        \
    const int row = (MT) * 16 + khalf * 8 + j;                         \
    if (row < N_PT) ob[(size_t)row * HW] = (ACC)[j];                   \
  }
  STORE_TILE(acc0, 0)
  STORE_TILE(acc1, 1)
  STORE_TILE(acc2, 2)
  STORE_TILE(acc3, 3)
#undef STORE_TILE
}

// ---------------------------------------------------------------------------
extern "C" void kernel_launch(void* const* d_in, const int* in_sizes, int n_in,
                              void* d_out, int out_size, void* d_ws, size_t ws_size,
                              hipStream_t stream) {
  const float* x        = (const float*)d_in[0];  // (8, 64, 67200)
  const float* idx_feat = (const float*)d_in[1];  // (8, 256, 50, 1)
  const float* weight   = (const float*)d_in[2];  // (64, 256)
  const float* bias     = (const float*)d_in[3];  // (64,)
  (void)in_sizes; (void)n_in; (void)out_size; (void)ws_size;

  float* out = (float*)d_out;                     // 8*50*67200 inst_pred + 1 loss
  float* kw  = (float*)d_ws;                      // kernels (8,50,64) fp32

  mw_kernel<<<dim3(NB), 256, 0, stream>>>(idx_feat, weight, bias, kw);
  con_loss_kernel<<<dim3(1), 256, 0, stream>>>(kw, out + (size_t)NB * N_PT * HW);
  inst_pred_wmma<<<dim3(HW / PIX_PER_BLOCK, NB), 256, 0, stream>>>(x, kw, out);
}